// TransE_57681410785658
// MI455X (gfx1250) — compile-verified
//
#include <hip/hip_runtime.h>
#include <math.h>

// TransE margin-ranking loss on MI455X (gfx1250, wave32).
// Strategy: gather-bound kernel; one wave handles 16 triples, per-lane f32
// squared-diff partials are reduced across lanes/K-chunks with a single
// V_WMMA_F32_16X16X4_F32 (B = ones) per pos/neg side. Deterministic two-stage
// reduction (no float atomics).

typedef __attribute__((ext_vector_type(2))) float v2f;
typedef __attribute__((ext_vector_type(8))) float v8f;

#define DIM      256
#define MARGIN   1.0f
#define INV_B    (1.0f / 4096.0f)

// Sum over 64 consecutive dims of (h + r - t)^2, vectorized 128-bit loads.
__device__ __forceinline__ float sq64(const float* __restrict__ h,
                                      const float* __restrict__ r,
                                      const float* __restrict__ t) {
  const float4* __restrict__ h4 = (const float4*)h;
  const float4* __restrict__ r4 = (const float4*)r;
  const float4* __restrict__ t4 = (const float4*)t;
  float acc = 0.0f;
#pragma unroll 4
  for (int j = 0; j < 16; ++j) {
    float4 a = h4[j];
    float4 b = r4[j];
    float4 c = t4[j];
    float dx = a.x + b.x - c.x;
    float dy = a.y + b.y - c.y;
    float dz = a.z + b.z - c.z;
    float dw = a.w + b.w - c.w;
    acc = fmaf(dx, dx, acc);
    acc = fmaf(dy, dy, acc);
    acc = fmaf(dz, dz, acc);
    acc = fmaf(dw, dw, acc);
  }
  return acc;
}

// One wave (32 lanes) per block; block g handles triples [16g, 16g+16).
// Lane L: triple-row m = L&15, K-half kh = L>>4.
// A-matrix (16x4 f32): lane(m, kh) holds K=2kh (VGPR0) and K=2kh+1 (VGPR1)
// = partial sums over dims [128kh,128kh+64) and [128kh+64,128kh+128).
// D = A x ones + 0  =>  D[m][n] = ||d_m||^2 for every column n.
__global__ void __launch_bounds__(32)
transe_wmma_kernel(const float* __restrict__ ent, const float* __restrict__ rel,
                   const int* __restrict__ ph, const int* __restrict__ pr,
                   const int* __restrict__ pt, const int* __restrict__ nh,
                   const int* __restrict__ nr, const int* __restrict__ nt,
                   float* __restrict__ partials) {
  const int lane = (int)(threadIdx.x & 31);
  const int m    = lane & 15;
  const int kh   = lane >> 4;
  const int trip = (int)blockIdx.x * 16 + m;
  const int base = kh * 128;

  const float* __restrict__ h1 = ent + (size_t)ph[trip] * DIM;
  const float* __restrict__ r1 = rel + (size_t)pr[trip] * DIM;
  const float* __restrict__ t1 = ent + (size_t)pt[trip] * DIM;
  const float* __restrict__ h2 = ent + (size_t)nh[trip] * DIM;
  const float* __restrict__ r2 = rel + (size_t)nr[trip] * DIM;
  const float* __restrict__ t2 = ent + (size_t)nt[trip] * DIM;

  v2f Ap, An;
  Ap.x = sq64(h1 + base,      r1 + base,      t1 + base);
  Ap.y = sq64(h1 + base + 64, r1 + base + 64, t1 + base + 64);
  An.x = sq64(h2 + base,      r2 + base,      t2 + base);
  An.y = sq64(h2 + base + 64, r2 + base + 64, t2 + base + 64);

  v2f Bones;
  Bones.x = 1.0f;
  Bones.y = 1.0f;
  v8f Zero = {};

  // 8 args: (neg_a, A, neg_b, B, c_mod, C, reuse_a, reuse_b)
  v8f Dp = __builtin_amdgcn_wmma_f32_16x16x4_f32(false, Ap, false, Bones,
                                                 (short)0, Zero, false, false);
  v8f Dn = __builtin_amdgcn_wmma_f32_16x16x4_f32(false, An, false, Bones,
                                                 (short)0, Zero, false, false);

  // D VGPR j: lanes 0-15 -> row j, lanes 16-31 -> row j+8. Each lane sums the
  // hinges of its 8 rows; result identical within each 16-lane half.
  float s = 0.0f;
#pragma unroll
  for (int j = 0; j < 8; ++j) {
    float hinge = (MARGIN + sqrtf(Dp[j]) - sqrtf(Dn[j])) * INV_B;
    s += fmaxf(hinge, 0.0f);
  }
  float lo = __shfl(s, 0, 32);   // rows 0..7
  float hi = __shfl(s, 16, 32);  // rows 8..15
  if (lane == 0) partials[blockIdx.x] = lo + hi;
}

// Deterministic fixed-tree reduction of up to 256 partials into one scalar.
__global__ void __launch_bounds__(256)
reduce_partials(const float* __restrict__ partials, float* __restrict__ out,
                int n) {
  __shared__ float buf[256];
  const int t = (int)threadIdx.x;
  buf[t] = (t < n) ? partials[t] : 0.0f;
  __syncthreads();
#pragma unroll
  for (int s = 128; s > 0; s >>= 1) {
    if (t < s) buf[t] += buf[t + s];
    __syncthreads();
  }
  if (t == 0) out[0] = buf[0];
}

extern "C" void kernel_launch(void* const* d_in, const int* in_sizes, int n_in,
                              void* d_out, int out_size, void* d_ws,
                              size_t ws_size, hipStream_t stream) {
  const float* ent = (const float*)d_in[0];
  const float* rel = (const float*)d_in[1];
  const int* ph = (const int*)d_in[2];
  const int* pr = (const int*)d_in[3];
  const int* pt = (const int*)d_in[4];
  const int* nh = (const int*)d_in[5];
  const int* nr = (const int*)d_in[6];
  const int* nt = (const int*)d_in[7];

  const int batch  = in_sizes[2];      // 4096
  const int groups = batch / 16;       // 256 wave-groups of 16 triples

  float* partials = (float*)d_ws;

  transe_wmma_kernel<<<groups, 32, 0, stream>>>(ent, rel, ph, pr, pt, nh, nr,
                                                nt, partials);
  reduce_partials<<<1, 256, 0, stream>>>(partials, (float*)d_out, groups);
}